// aspect_rating_1_17875653886124
// MI455X (gfx1250) — compile-verified
//
#include <hip/hip_runtime.h>
#include <hip/hip_bf16.h>
#include <math.h>

#define R_   2048
#define L_   256
#define D_   200
#define A_   20
#define NEG_ 5
#define B_   1024
#define NNZ_ 8192

typedef float v2f __attribute__((ext_vector_type(2)));
typedef float v8f __attribute__((ext_vector_type(8)));

// ---------------------------------------------------------------------------
// Kernel 1: V[r,e] = sum_d Y[r,d] * M[d,e]   (2048 x 200 x 200 f32 GEMM)
// One wave32 computes one 16x16 tile via V_WMMA_F32_16X16X4_F32.
// A (16x4): lane m=lane&15; a = {K(kh*2), K(kh*2+1)}, kh=lane>>4  (one b64 load)
// B (4x16): lane n=lane&15; b = {row(kh*2), row(kh*2+1)} of M
// D (16x16): vgpr j holds M=j+8*kh, N=lane&15
// Edge tile (N=200 % 16): unconditional clamped loads * 0/1 mask -> EXEC stays
// all-ones through the whole loop (WMMA requirement), no exec save/restore.
// ---------------------------------------------------------------------------
__global__ void gemm_v_wmma(const float* __restrict__ Y,
                            const float* __restrict__ M,
                            float* __restrict__ V) {
  const int lane = threadIdx.x & 31;
  const int wave = threadIdx.x >> 5;
  const int tile = blockIdx.x * (blockDim.x >> 5) + wave;
  const int NT = (D_ + 15) / 16;              // 13 column tiles
  const int tm = tile / NT;                   // 0..127
  const int tn = tile % NT;                   // 0..12
  const int m  = lane & 15;
  const int kh = lane >> 4;                   // 0 or 1
  const int arow = tm * 16 + m;
  const int bcol = tn * 16 + m;               // n-bits == m-bits of lane
  const bool  bok  = (bcol < D_);
  const int   bcc  = bok ? bcol : (D_ - 1);   // clamped (always in-bounds)
  const float bmsk = bok ? 1.f : 0.f;

  const float* __restrict__ Arow = Y + (size_t)arow * D_ + kh * 2;
  const float* __restrict__ Bcol = M + (size_t)(kh * 2) * D_ + bcc;

  v8f c = {0.f, 0.f, 0.f, 0.f, 0.f, 0.f, 0.f, 0.f};
#pragma unroll 2
  for (int k = 0; k < D_; k += 4) {
    v2f a = *(const v2f*)(Arow + k);          // single 8B load (even offset)
    v2f b;
    b.x = Bcol[(size_t)k * D_]        * bmsk;
    b.y = Bcol[(size_t)(k + 1) * D_]  * bmsk;
    c = __builtin_amdgcn_wmma_f32_16x16x4_f32(false, a, false, b,
                                              (short)0, c, false, false);
  }
  if (bok) {
#pragma unroll
    for (int j = 0; j < 8; ++j) {
      int row = tm * 16 + j + kh * 8;
      V[(size_t)row * D_ + bcol] = c[j];
    }
  }
}

// ---------------------------------------------------------------------------
// Kernel 2: per-review attention + aspect head. One block (256 thr) per r.
// dx[l] = emb[idx[r,l]] . v[r];  ax = softmax_l(dx);  z = sum_l ax[l]*emb[.]
// p = softmax_A(z @ Ww^T + Wb);  rs = p @ Tw^T;  c1 = cos(rs, z)
// ---------------------------------------------------------------------------
__global__ void review_kernel(const int* __restrict__ idx,
                              const float* __restrict__ Vv,
                              const float* __restrict__ emb,
                              const float* __restrict__ Ww,
                              const float* __restrict__ Wb,
                              const float* __restrict__ Tw,
                              float* __restrict__ rs_out,
                              float* __restrict__ c1_out,
                              float* __restrict__ rsninv_out) {
  __shared__ __align__(16) float v_sh[D_];
  __shared__ float ax[L_];
  __shared__ int   id_sh[L_];
  __shared__ float red[L_];
  __shared__ float z_sh[D_];
  __shared__ float rs_sh[D_];
  __shared__ float p_sh[A_];
  __shared__ float sc[2];

  const int t = threadIdx.x;      // 0..255
  const int r = blockIdx.x;

  id_sh[t] = idx[(size_t)r * L_ + t];
  if (t < D_) v_sh[t] = Vv[(size_t)r * D_ + t];
  __syncthreads();

  // attention logit for token t
  const float4* e4 = (const float4*)(emb + (size_t)id_sh[t] * D_);
  const float4* v4 = (const float4*)v_sh;
  float s = 0.f;
#pragma unroll 5
  for (int i = 0; i < D_ / 4; ++i) {
    float4 ev = e4[i]; float4 vv = v4[i];
    s += ev.x * vv.x + ev.y * vv.y + ev.z * vv.z + ev.w * vv.w;
  }
  // softmax over L
  red[t] = s; __syncthreads();
  for (int off = L_ / 2; off > 0; off >>= 1) {
    if (t < off) red[t] = fmaxf(red[t], red[t + off]);
    __syncthreads();
  }
  float mx = red[0]; __syncthreads();
  float ev_ = __expf(s - mx);
  red[t] = ev_; __syncthreads();
  for (int off = L_ / 2; off > 0; off >>= 1) {
    if (t < off) red[t] += red[t + off];
    __syncthreads();
  }
  float denom = red[0]; __syncthreads();
  ax[t] = ev_ / denom;
  __syncthreads();

  // z[d] = sum_l ax[l] * emb[id[l]][d]   (coalesced over d)
  if (t < D_) {
    float z = 0.f;
    for (int l = 0; l < L_; ++l)
      z += ax[l] * emb[(size_t)id_sh[l] * D_ + t];
    z_sh[t] = z;
  }
  __syncthreads();

  // aspect logits (A=20)
  if (t < A_) {
    float pl = Wb[t];
    const float* w = Ww + (size_t)t * D_;
    for (int d = 0; d < D_; ++d) pl += w[d] * z_sh[d];
    p_sh[t] = pl;
  }
  __syncthreads();
  if (t == 0) {
    float pm = p_sh[0];
    for (int a = 1; a < A_; ++a) pm = fmaxf(pm, p_sh[a]);
    float ps = 0.f;
    for (int a = 0; a < A_; ++a) { float pe = __expf(p_sh[a] - pm); p_sh[a] = pe; ps += pe; }
    for (int a = 0; a < A_; ++a) p_sh[a] /= ps;
  }
  __syncthreads();

  // rs[d] = sum_a p[a] * Tw[d,a]
  if (t < D_) {
    float rs = 0.f;
#pragma unroll
    for (int a = 0; a < A_; ++a) rs += p_sh[a] * Tw[(size_t)t * A_ + a];
    rs_sh[t] = rs;
    rs_out[(size_t)r * D_ + t] = rs;
  }
  __syncthreads();

  // ||z||^2, ||rs||^2, z.rs
  float zz = 0.f, rr = 0.f, zr = 0.f;
  if (t < D_) { float z = z_sh[t], q = rs_sh[t]; zz = z * z; rr = q * q; zr = z * q; }
  red[t] = zz; __syncthreads();
  for (int off = L_ / 2; off > 0; off >>= 1) { if (t < off) red[t] += red[t + off]; __syncthreads(); }
  if (t == 0) sc[0] = red[0]; __syncthreads();
  red[t] = rr; __syncthreads();
  for (int off = L_ / 2; off > 0; off >>= 1) { if (t < off) red[t] += red[t + off]; __syncthreads(); }
  if (t == 0) sc[1] = red[0]; __syncthreads();
  red[t] = zr; __syncthreads();
  for (int off = L_ / 2; off > 0; off >>= 1) { if (t < off) red[t] += red[t + off]; __syncthreads(); }
  if (t == 0) {
    float nz = fmaxf(sqrtf(sc[0]), 1e-12f);
    float nr = fmaxf(sqrtf(sc[1]), 1e-12f);
    c1_out[r]     = red[0] / (nz * nr);
    rsninv_out[r] = 1.f / nr;
  }
}

// ---------------------------------------------------------------------------
// Kernel 3: margin loss per (r, neg). One wave32 per pair.
// ---------------------------------------------------------------------------
__global__ void neg_loss_kernel(const float* __restrict__ zn,
                                const float* __restrict__ rs,
                                const float* __restrict__ c1,
                                const float* __restrict__ rsninv,
                                float* __restrict__ abae_out) {
  const int lane = threadIdx.x & 31;
  const int wave = threadIdx.x >> 5;
  const int pair = blockIdx.x * (blockDim.x >> 5) + wave;   // 0..R*NEG-1
  const int r = pair / NEG_;
  const float* z  = zn + (size_t)pair * D_;
  const float* rr = rs + (size_t)r * D_;
  float d = 0.f, nn = 0.f;
  for (int i = lane; i < D_; i += 32) {
    float zv = z[i];
    d  += zv * rr[i];
    nn += zv * zv;
  }
  for (int off = 16; off > 0; off >>= 1) {
    d  += __shfl_down(d,  off, 32);
    nn += __shfl_down(nn, off, 32);
  }
  if (lane == 0) {
    float c2 = d * rsninv[r] / fmaxf(sqrtf(nn), 1e-12f);
    abae_out[pair] = fmaxf(0.f, -(c1[r] - c2) + 1.0f);
  }
}

// ---------------------------------------------------------------------------
// Kernel 4: orthogonality regularizer  U = mean((Tn^T Tn - I)^2), single block
// ---------------------------------------------------------------------------
__global__ void uloss_kernel(const float* __restrict__ Tw, float* __restrict__ U_out) {
  __shared__ float cninv[A_];
  __shared__ float red[256];
  const int t = threadIdx.x;
  if (t < A_) {
    float s = 0.f;
    for (int d = 0; d < D_; ++d) { float v = Tw[(size_t)d * A_ + t]; s += v * v; }
    cninv[t] = 1.f / fmaxf(sqrtf(s), 1e-12f);
  }
  __syncthreads();
  float acc = 0.f;
  for (int p = t; p < A_ * A_; p += 256) {
    int i = p / A_, j = p % A_;
    float g = 0.f;
    for (int d = 0; d < D_; ++d) g += Tw[(size_t)d * A_ + i] * Tw[(size_t)d * A_ + j];
    g *= cninv[i] * cninv[j];
    float diff = g - ((i == j) ? 1.f : 0.f);
    acc += diff * diff;
  }
  red[t] = acc; __syncthreads();
  for (int off = 128; off > 0; off >>= 1) { if (t < off) red[t] += red[t + off]; __syncthreads(); }
  if (t == 0) U_out[0] = red[0] / (float)(A_ * A_);
}

// ---------------------------------------------------------------------------
// Zero-fill (graph-safe replacement for memset; atomics need fresh zeros)
// ---------------------------------------------------------------------------
__global__ void zero_kernel(float* __restrict__ p, int n) {
  int i = blockIdx.x * blockDim.x + threadIdx.x;
  if (i < n) p[i] = 0.f;
}

// ---------------------------------------------------------------------------
// Kernel 5: COO scatter-add for user/item embeddings. One block per nnz entry.
// ---------------------------------------------------------------------------
__global__ void scatter_kernel(const int* __restrict__ u_idx, const float* __restrict__ u_val,
                               const int* __restrict__ i_idx, const float* __restrict__ i_val,
                               const float* __restrict__ rs,
                               float* __restrict__ ue, float* __restrict__ ie) {
  const int e = blockIdx.x;
  const int d = threadIdx.x;
  if (d >= D_) return;
  {
    int row = u_idx[e];
    int col = u_idx[NNZ_ + e];
    atomicAdd(&ue[(size_t)row * D_ + d], u_val[e] * rs[(size_t)col * D_ + d]);
  }
  {
    int row = i_idx[e];
    int col = i_idx[NNZ_ + e];
    atomicAdd(&ie[(size_t)row * D_ + d], i_val[e] * rs[(size_t)col * D_ + d]);
  }
}

// ---------------------------------------------------------------------------
// Kernel 6: prediction + squared rating loss. One wave32 per b.
// ---------------------------------------------------------------------------
__global__ void rating_kernel(const float* __restrict__ ue, const float* __restrict__ ie,
                              const float* __restrict__ label,
                              float* __restrict__ pred_out, float* __restrict__ rl_out) {
  const int lane = threadIdx.x & 31;
  const int wave = threadIdx.x >> 5;
  const int b = blockIdx.x * (blockDim.x >> 5) + wave;
  const float* u = ue + (size_t)b * D_;
  const float* v = ie + (size_t)b * D_;
  float s = 0.f;
  for (int i = lane; i < D_; i += 32) s += u[i] * v[i];
  for (int off = 16; off > 0; off >>= 1) s += __shfl_down(s, off, 32);
  if (lane == 0) {
    float pred = s + 3.5f;
    pred_out[b] = pred;
    float diff = pred - label[b];
    rl_out[b] = diff * diff;
  }
}

// ---------------------------------------------------------------------------
// Kernel 7: deterministic final reduction -> obj scalar
// ---------------------------------------------------------------------------
__global__ void finalize_kernel(const float* __restrict__ rl,
                                const float* __restrict__ abae,
                                const float* __restrict__ U,
                                float* __restrict__ obj) {
  __shared__ float red[256];
  const int t = threadIdx.x;
  float j = 0.f;
  for (int i = t; i < R_ * NEG_; i += 256) j += abae[i];
  float rsum = 0.f;
  for (int i = t; i < B_; i += 256) rsum += rl[i];
  red[t] = j; __syncthreads();
  for (int off = 128; off > 0; off >>= 1) { if (t < off) red[t] += red[t + off]; __syncthreads(); }
  float jtot = red[0]; __syncthreads();
  red[t] = rsum; __syncthreads();
  for (int off = 128; off > 0; off >>= 1) { if (t < off) red[t] += red[t + off]; __syncthreads(); }
  if (t == 0)
    obj[0] = red[0] / (float)B_ + U[0] + jtot / (float)(R_ * NEG_);
}

// ---------------------------------------------------------------------------
extern "C" void kernel_launch(void* const* d_in, const int* in_sizes, int n_in,
                              void* d_out, int out_size, void* d_ws, size_t ws_size,
                              hipStream_t stream) {
  const int*   hist   = (const int*)  d_in[0];   // (R, L)
  const float* y_pos  = (const float*)d_in[1];   // (R, D)
  const float* z_neg  = (const float*)d_in[2];   // (R*NEG, D)
  const float* label  = (const float*)d_in[5];   // (B,)
  const int*   u_idx  = (const int*)  d_in[6];   // (2, NNZ)
  const float* u_val  = (const float*)d_in[7];   // (NNZ,)
  const int*   i_idx  = (const int*)  d_in[8];   // (2, NNZ)
  const float* i_val  = (const float*)d_in[9];   // (NNZ,)
  const float* emb    = (const float*)d_in[10];  // (VOCAB, D)
  const float* M_w    = (const float*)d_in[11];  // (D, D)
  const float* W_w    = (const float*)d_in[12];  // (A, D)
  const float* W_b    = (const float*)d_in[13];  // (A,)
  const float* T_w    = (const float*)d_in[14];  // (D, A)

  float* ws = (float*)d_ws;
  float* V      = ws;                                  // R*D
  float* rs     = V      + (size_t)R_ * D_;            // R*D
  float* c1     = rs     + (size_t)R_ * D_;            // R
  float* rsninv = c1     + R_;                         // R
  float* ue     = rsninv + R_;                         // B*D
  float* ie     = ue     + (size_t)B_ * D_;            // B*D
  float* U      = ie     + (size_t)B_ * D_;            // 1

  float* out        = (float*)d_out;
  float* out_obj    = out;                 // [0]
  float* out_rl     = out + 1;             // [1 .. 1024]
  float* out_abae   = out + 1 + B_;        // [1025 .. 11264]
  float* out_pred   = out + 1 + B_ + R_ * NEG_;  // [11265 .. 12288]

  // 0) zero the scatter accumulators (ue, ie contiguous)
  {
    int n = 2 * B_ * D_;
    zero_kernel<<<(n + 255) / 256, 256, 0, stream>>>(ue, n);
  }
  // 1) V = Y @ M  (WMMA f32)
  {
    int tiles = (R_ / 16) * ((D_ + 15) / 16);          // 128 * 13 = 1664
    gemm_v_wmma<<<tiles / 8, 256, 0, stream>>>(y_pos, M_w, V);
  }
  // 2) attention + aspect head per review
  review_kernel<<<R_, 256, 0, stream>>>(hist, V, emb, W_w, W_b, T_w, rs, c1, rsninv);
  // 3) margin losses
  neg_loss_kernel<<<(R_ * NEG_) / 8, 256, 0, stream>>>(z_neg, rs, c1, rsninv, out_abae);
  // 4) orthogonality regularizer
  uloss_kernel<<<1, 256, 0, stream>>>(T_w, U);
  // 5) sparse scatter-add
  scatter_kernel<<<NNZ_, 256, 0, stream>>>(u_idx, u_val, i_idx, i_val, rs, ue, ie);
  // 6) prediction + rating loss
  rating_kernel<<<B_ / 8, 256, 0, stream>>>(ue, ie, label, out_pred, out_rl);
  // 7) objective scalar (deterministic reduction)
  finalize_kernel<<<1, 256, 0, stream>>>(out_rl, out_abae, U, out_obj);
}